// Task5Model_73280732005211
// MI455X (gfx1250) — compile-verified
//
#include <hip/hip_runtime.h>

typedef __attribute__((ext_vector_type(2))) float v2f;
typedef __attribute__((ext_vector_type(8))) float v8f;

#define T_LEN 262144
#define BATCH 4

// ---------------------------------------------------------------------------
// Fast device helpers (serial scan is latency-bound: keep chains short)
// ---------------------------------------------------------------------------
__device__ __forceinline__ float rcp_fast(float x) {
#if __has_builtin(__builtin_amdgcn_rcpf)
    return __builtin_amdgcn_rcpf(x);
#else
    return __fdividef(1.0f, x);
#endif
}
__device__ __forceinline__ float sigm(float x) {
    return rcp_fast(1.0f + __expf(-x));
}
__device__ __forceinline__ float tanh_s(float x) {   // tanh via 2*sigmoid(2x)-1
    return fmaf(2.0f, sigm(2.0f * x), -1.0f);
}

// ---------------------------------------------------------------------------
// Phase 1: xg[(t*B+b)*8 + g] = x[t,b] * Wih0[g] + (bih0[g] + bhh0[g])
// V_WMMA_F32_16X16X4_F32 per 16 flat (t,b) rows:
//   A: M=16 rows, K0 = x value, K1..K3 = 0
//   B: K0 row = Wih0[N] for N<8, everything else 0   (loop-invariant)
//   C: bias broadcast down rows (per-column N)        (loop-invariant)
// Unrolled x4 so the four independent x loads pipeline under one wait and
// the four WMMAs + store clause issue back to back.
// ---------------------------------------------------------------------------
__global__ void lstm_xgemm_wmma(const float* __restrict__ x,
                                const float* __restrict__ Wih0,
                                const float* __restrict__ bih0,
                                const float* __restrict__ bhh0,
                                float* __restrict__ xg) {
    const int lane   = threadIdx.x & 31;
    const int wid    = (blockIdx.x * blockDim.x + threadIdx.x) >> 5;
    const int nwaves = (gridDim.x * blockDim.x) >> 5;
    const int col    = lane & 15;

    // B operand: VGPR0 = {K=0 | K=2} rows, VGPR1 = {K=1 | K=3} rows.
    v2f bmat;
    bmat.x = (lane < 8) ? Wih0[lane] : 0.0f;
    bmat.y = 0.0f;

    // C operand: bias depends only on column N = lane&15 (broadcast over rows)
    const float bias = (col < 8) ? (bih0[col] + bhh0[col]) : 0.0f;
    v8f cmat;
#pragma unroll
    for (int r = 0; r < 8; ++r) cmat[r] = bias;

    const int ngroups = (T_LEN * BATCH) / 64;   // 16384 groups of 4 tiles
    for (int g = wid; g < ngroups; g += nwaves) {
        const int base = g * 64;                // 64 consecutive (t,b) rows

        float xv[4];
#pragma unroll
        for (int u = 0; u < 4; ++u)             // 4 independent coalesced loads
            xv[u] = x[base + u * 16 + col];

        v8f d[4];
#pragma unroll
        for (int u = 0; u < 4; ++u) {
            v2f amat;
            amat.x = (lane < 16) ? xv[u] : 0.0f;   // K0 / K2
            amat.y = 0.0f;                         // K1 / K3
            d[u] = __builtin_amdgcn_wmma_f32_16x16x4_f32(
                false, amat, false, bmat, (short)0, cmat, false, false);
        }

        // D: VGPR r -> row (tile_base + r) lanes 0..15, (tile_base + 8 + r) lanes 16..31
        if (col < 8) {
#pragma unroll
            for (int u = 0; u < 4; ++u) {
                const int row0 = base + u * 16 + ((lane >> 4) << 3);
#pragma unroll
                for (int r = 0; r < 8; ++r)
                    xg[(row0 + r) * 8 + col] = d[u][r];
            }
        }
    }
}

// ---------------------------------------------------------------------------
// Phase 2: single-wave sequential scan, both layers pipelined.
// Lane L: b = L>>3 (batch), slot = L&7 (gate vector [i0 i1 f0 f1 g0 g1 o0 o1])
// State (h,c per batch) is replicated across the 8 lanes of its batch, so
// 8 ds_bpermute gathers per layer per step rebuild replication for free.
// ---------------------------------------------------------------------------
template <bool USE_XG>
__global__ void __launch_bounds__(32)
lstm_scan(const float* __restrict__ x,     // [T,B] (IN==1)
          const float* __restrict__ xg,    // [T*B, 8] precomputed (or null)
          const float* __restrict__ h0,    // [2,B,H]
          const float* __restrict__ c0,    // [2,B,H]
          const float* __restrict__ Wih0, const float* __restrict__ Whh0,
          const float* __restrict__ bih0, const float* __restrict__ bhh0,
          const float* __restrict__ Wih1, const float* __restrict__ Whh1,
          const float* __restrict__ bih1, const float* __restrict__ bhh1,
          float* __restrict__ out)         // [T,B,H]
{
    const int L     = threadIdx.x;
    const int b     = L >> 3;
    const int slot  = L & 7;
    const int gtype = slot >> 1;          // 0:i 1:f 2:g 3:o
    const int hh    = slot & 1;
    const int gbase = b << 3;

    // gate activation = as * sigmoid(am * pre) + ao  (tanh for the g-gate)
    const float am = (gtype == 2) ? 2.0f : 1.0f;
    const float as = (gtype == 2) ? 2.0f : 1.0f;
    const float ao = (gtype == 2) ? -1.0f : 0.0f;

    // per-lane (per gate-slot) weights
    const float w1h0 = Whh0[slot * 2 + 0], w1h1 = Whh0[slot * 2 + 1];
    const float w2i0 = Wih1[slot * 2 + 0], w2i1 = Wih1[slot * 2 + 1];
    const float w2h0 = Whh1[slot * 2 + 0], w2h1 = Whh1[slot * 2 + 1];
    const float b2   = bih1[slot] + bhh1[slot];
    float b1 = 0.0f, w1i = 0.0f;
    if (!USE_XG) { b1 = bih0[slot] + bhh0[slot]; w1i = Wih0[slot]; }

    // replicated state: h/c for layer1 and layer2, this lane's batch
    float h1_0 = h0[b * 2 + 0],     h1_1 = h0[b * 2 + 1];
    float c1_0 = c0[b * 2 + 0],     c1_1 = c0[b * 2 + 1];
    float h2_0 = h0[8 + b * 2 + 0], h2_1 = h0[8 + b * 2 + 1];
    float c2_0 = c0[8 + b * 2 + 0], c2_1 = c0[8 + b * 2 + 1];

    // layer-2 step (consumes y1 = (y0v,y1v), updates state refs in place)
    auto layer2_step = [&](float y0v, float y1v,
                           float& pc0, float& pc1, float& ph0, float& ph1) {
        float pre2 = b2;
        pre2 = fmaf(y0v, w2i0, pre2);
        pre2 = fmaf(y1v, w2i1, pre2);
        pre2 = fmaf(ph0, w2h0, pre2);
        pre2 = fmaf(ph1, w2h1, pre2);
        const float a2 = fmaf(as, sigm(am * pre2), ao);
        const float i0 = __shfl(a2, gbase + 0, 32), i1 = __shfl(a2, gbase + 1, 32);
        const float f0 = __shfl(a2, gbase + 2, 32), f1 = __shfl(a2, gbase + 3, 32);
        const float g0 = __shfl(a2, gbase + 4, 32), g1 = __shfl(a2, gbase + 5, 32);
        const float o0 = __shfl(a2, gbase + 6, 32), o1 = __shfl(a2, gbase + 7, 32);
        pc0 = fmaf(f0, pc0, i0 * g0);
        pc1 = fmaf(f1, pc1, i1 * g1);
        ph0 = o0 * tanh_s(pc0);
        ph1 = o1 * tanh_s(pc1);
    };

    for (int t = 0; t < T_LEN; ++t) {
        // ---- layer-1 pre-activation, step t (reads OLD h1)
        float pre1 = USE_XG ? xg[t * 32 + L]               // coalesced 128B/step
                            : fmaf(x[t * 4 + b], w1i, b1);
        pre1 = fmaf(h1_0, w1h0, pre1);
        pre1 = fmaf(h1_1, w1h1, pre1);
        const float a1 = fmaf(as, sigm(am * pre1), ao);

        // ---- layer-2 step t-1, pipelined: input y1[t-1] == current (old) h1
        float tc0 = c2_0, tc1 = c2_1, th0 = h2_0, th1 = h2_1;
        layer2_step(h1_0, h1_1, tc0, tc1, th0, th1);
        if (t > 0) {                                   // wave-uniform branch
            c2_0 = tc0; c2_1 = tc1; h2_0 = th0; h2_1 = th1;
            if (slot < 2)
                out[(t - 1) * 8 + b * 2 + hh] = hh ? th1 : th0;
        }

        // ---- commit layer-1 state (gather i/f/g/o for this batch)
        const float i0 = __shfl(a1, gbase + 0, 32), i1 = __shfl(a1, gbase + 1, 32);
        const float f0 = __shfl(a1, gbase + 2, 32), f1 = __shfl(a1, gbase + 3, 32);
        const float g0 = __shfl(a1, gbase + 4, 32), g1 = __shfl(a1, gbase + 5, 32);
        const float o0 = __shfl(a1, gbase + 6, 32), o1 = __shfl(a1, gbase + 7, 32);
        c1_0 = fmaf(f0, c1_0, i0 * g0);
        c1_1 = fmaf(f1, c1_1, i1 * g1);
        h1_0 = o0 * tanh_s(c1_0);
        h1_1 = o1 * tanh_s(c1_1);
    }

    // epilogue: layer-2 step T-1 (input y1[T-1] = final h1)
    layer2_step(h1_0, h1_1, c2_0, c2_1, h2_0, h2_1);
    if (slot < 2)
        out[(T_LEN - 1) * 8 + b * 2 + hh] = hh ? h2_1 : h2_0;
}

// ---------------------------------------------------------------------------
extern "C" void kernel_launch(void* const* d_in, const int* in_sizes, int n_in,
                              void* d_out, int out_size, void* d_ws, size_t ws_size,
                              hipStream_t stream) {
    (void)in_sizes; (void)n_in; (void)out_size;
    const float* x    = (const float*)d_in[0];
    const float* h0   = (const float*)d_in[1];
    const float* c0   = (const float*)d_in[2];
    const float* Wih0 = (const float*)d_in[3];
    const float* Whh0 = (const float*)d_in[4];
    const float* bih0 = (const float*)d_in[5];
    const float* bhh0 = (const float*)d_in[6];
    const float* Wih1 = (const float*)d_in[7];
    const float* Whh1 = (const float*)d_in[8];
    const float* bih1 = (const float*)d_in[9];
    const float* bhh1 = (const float*)d_in[10];
    float* out = (float*)d_out;

    const size_t xg_bytes = (size_t)T_LEN * BATCH * 8 * sizeof(float);  // 32 MB
    if (ws_size >= xg_bytes) {
        float* xg = (float*)d_ws;
        lstm_xgemm_wmma<<<256, 256, 0, stream>>>(x, Wih0, bih0, bhh0, xg);
        lstm_scan<true><<<1, 32, 0, stream>>>(x, xg, h0, c0,
                                              Wih0, Whh0, bih0, bhh0,
                                              Wih1, Whh1, bih1, bhh1, out);
    } else {
        lstm_scan<false><<<1, 32, 0, stream>>>(x, nullptr, h0, c0,
                                               Wih0, Whh0, bih0, bhh0,
                                               Wih1, Whh1, bih1, bhh1, out);
    }
}